// StructDec_44135083933973
// MI455X (gfx1250) — compile-verified
//
#include <hip/hip_runtime.h>

// GCNConv(64,64, self-loops, sym-norm) + bias + ReLU for MI455X (gfx1250).
//
// Roofline: edge scatter dominates (~820 MB of L2-resident gather+atomic
// traffic ≈ 35us at 23.3 TB/s; accumulator (25.6MB) and features (25.6MB)
// both live in the 192MB L2). The 100k x 64 x 64 GEMM (0.82 GFLOP) is done
// with the fp32 WMMA path (v_wmma_f32_16x16x4_f32) since the reference is
// fp32 end-to-end.

typedef __attribute__((ext_vector_type(2))) float v2f;
typedef __attribute__((ext_vector_type(4))) float v4f;
typedef __attribute__((ext_vector_type(8))) float v8f;

#define DIM 64

// --- pass 1: zero degree counters -------------------------------------------
__global__ void k_zero(unsigned int* __restrict__ deg, int N) {
  int i = blockIdx.x * blockDim.x + threadIdx.x;
  if (i < N) deg[i] = 0u;
}

// --- pass 2: in-degree (dst side, unit weights) -----------------------------
__global__ void k_deg(const int* __restrict__ ei, unsigned int* __restrict__ deg,
                      int E) {
  int e = blockIdx.x * blockDim.x + threadIdx.x;
  if (e < E) atomicAdd(&deg[ei[E + e]], 1u);  // ei[1][e] = dst
}

// --- pass 3: dinv = rsqrt(deg + 1)  (+1 = self loop) ------------------------
__global__ void k_dinv(const unsigned int* __restrict__ deg,
                       float* __restrict__ dinv, int N) {
  int i = blockIdx.x * blockDim.x + threadIdx.x;
  if (i < N) dinv[i] = rsqrtf((float)(deg[i] + 1u));
}

// --- pass 4: x = z @ W^T via v_wmma_f32_16x16x4_f32 -------------------------
// One wave -> 16x64 tile. Writes xs = x*dinv[row] (edge-gather operand) and
// accum(=d_out) = x*dinv[row]^2 (self-loop term), fully overwriting d_out.
__global__ void __launch_bounds__(256) k_gemm(
    const float* __restrict__ z, const float* __restrict__ W,
    const float* __restrict__ dinv, float* __restrict__ xs,
    float* __restrict__ accum, int N) {
  __shared__ float Wl[DIM * DIM];  // 16 KB of the 320 KB WGP LDS
  {
    // cooperative, fully coalesced b128 staging of W into LDS
    const v4f* Wg4 = (const v4f*)W;
    v4f* Wl4 = (v4f*)Wl;
    int t = threadIdx.x;
#pragma unroll
    for (int i = 0; i < (DIM * DIM / 4) / 256; ++i)
      Wl4[t + i * 256] = Wg4[t + i * 256];
  }
  __syncthreads();

  const int wave = threadIdx.x >> 5;
  const int lane = threadIdx.x & 31;
  const int row0 = (blockIdx.x * 8 + wave) * 16;
  if (row0 >= N) return;  // wave-uniform: EXEC stays all-ones for WMMA

  // A-frag layout (32-bit A 16x4): lanes 0-15 -> K={0,1}, lanes 16-31 -> K={2,3}
  const int m  = lane & 15;
  const int kk = (lane >> 4) << 1;
  const float* zrow = z + (size_t)(row0 + m) * DIM;

  v8f c0 = {}, c1 = {}, c2 = {}, c3 = {};  // 4 independent N-tiles hide RAW
#pragma unroll
  for (int k0 = 0; k0 < DIM; k0 += 4) {
    v2f a = *(const v2f*)(zrow + k0 + kk);
    // B[k][n] = W[n][k]; same half-wave K split as A
    v2f b0 = *(const v2f*)(&Wl[(0  + m) * DIM + k0 + kk]);
    v2f b1 = *(const v2f*)(&Wl[(16 + m) * DIM + k0 + kk]);
    v2f b2 = *(const v2f*)(&Wl[(32 + m) * DIM + k0 + kk]);
    v2f b3 = *(const v2f*)(&Wl[(48 + m) * DIM + k0 + kk]);
    c0 = __builtin_amdgcn_wmma_f32_16x16x4_f32(false, a, false, b0, (short)0, c0, false, false);
    c1 = __builtin_amdgcn_wmma_f32_16x16x4_f32(false, a, false, b1, (short)0, c1, false, false);
    c2 = __builtin_amdgcn_wmma_f32_16x16x4_f32(false, a, false, b2, (short)0, c2, false, false);
    c3 = __builtin_amdgcn_wmma_f32_16x16x4_f32(false, a, false, b3, (short)0, c3, false, false);
  }

  // C/D layout: lanes 0-15 -> M=r, lanes 16-31 -> M=r+8; N = lane&15
  const int hi  = lane >> 4;
  const int col = lane & 15;
#pragma unroll
  for (int r = 0; r < 8; ++r) {
    int row = row0 + r + hi * 8;
    float di = dinv[row];
    size_t base = (size_t)row * DIM + col;
    float x0 = c0[r] * di, x1 = c1[r] * di, x2 = c2[r] * di, x3 = c3[r] * di;
    xs[base +  0] = x0;  accum[base +  0] = x0 * di;
    xs[base + 16] = x1;  accum[base + 16] = x1 * di;
    xs[base + 32] = x2;  accum[base + 32] = x2 * di;
    xs[base + 48] = x3;  accum[base + 48] = x3 * di;
  }
}

// --- pass 5: edge scatter: accum[dst] += xs[src] * dinv[dst] ----------------
// One wave per edge; lane handles 2 floats -> 256B coalesced gather per wave,
// 64 global_atomic_add_f32 per edge landing in L2.
__global__ void __launch_bounds__(256) k_edge(
    const int* __restrict__ ei, const float* __restrict__ dinv,
    const float* __restrict__ xs, float* __restrict__ accum, int E) {
  int e = blockIdx.x * 8 + (threadIdx.x >> 5);
  if (e >= E) return;
  int lane = threadIdx.x & 31;
  int src = ei[e];
  int dst = ei[E + e];
  float nd = dinv[dst];
  v2f v = *(const v2f*)(xs + (size_t)src * DIM + lane * 2);
  float* p = accum + (size_t)dst * DIM + lane * 2;
  unsafeAtomicAdd(p + 0, v.x * nd);
  unsafeAtomicAdd(p + 1, v.y * nd);
}

// --- pass 6: out = relu(out + b) in place -----------------------------------
__global__ void k_fin(float* __restrict__ out, const float* __restrict__ b,
                      long long total) {
  long long i = (long long)blockIdx.x * blockDim.x + threadIdx.x;
  if (i < total) {
    float v = out[i] + b[i & (DIM - 1)];
    out[i] = v > 0.0f ? v : 0.0f;
  }
}

extern "C" void kernel_launch(void* const* d_in, const int* in_sizes, int n_in,
                              void* d_out, int out_size, void* d_ws, size_t ws_size,
                              hipStream_t stream) {
  const float* z  = (const float*)d_in[0];  // [N,64]
  const int*   ei = (const int*)  d_in[1];  // [2,E] flat: src then dst
  const float* W  = (const float*)d_in[2];  // [64,64]
  const float* b  = (const float*)d_in[3];  // [64]
  float* out = (float*)d_out;               // [N,64] fp32

  const int N = in_sizes[0] / DIM;
  const int E = in_sizes[1] / 2;

  // workspace: xs[N*64] | deg[N] (u32) | dinv[N]  (~26.4 MB)
  float* xs = (float*)d_ws;
  unsigned int* deg = (unsigned int*)(xs + (size_t)N * DIM);
  float* dinv = (float*)(deg + N);

  const int T = 256;
  k_zero<<<(N + T - 1) / T, T, 0, stream>>>(deg, N);
  k_deg <<<(E + T - 1) / T, T, 0, stream>>>(ei, deg, E);
  k_dinv<<<(N + T - 1) / T, T, 0, stream>>>(deg, dinv, N);
  k_gemm<<<(N + 127) / 128, T, 0, stream>>>(z, W, dinv, xs, out, N);
  k_edge<<<(E + 7) / 8, T, 0, stream>>>(ei, dinv, xs, out, E);
  long long total = (long long)N * DIM;
  k_fin <<<(int)((total + T - 1) / T), T, 0, stream>>>(out, b, total);
}